// LinearAttention_738734375518
// MI455X (gfx1250) — compile-verified
//
#include <hip/hip_runtime.h>
#include <hip/hip_bf16.h>
#include <stdint.h>

typedef __attribute__((ext_vector_type(2))) float v2f;
typedef __attribute__((ext_vector_type(8))) float v8f;

#define HEADS   64          // b*h = 4*16
#define SEQ     8192
#define DIM     128
#define NSTEP   16          // staged sequence rows per async stage (phase 1)
#define CHUNK   512         // sequence rows per workgroup (phase 1)
#define NSTAGES (CHUNK / NSTEP)
#define P1_STR  136         // LDS row stride (floats); 2-row offset = 272 -> +16 banks (conflict-free halves)
#define P2_STR  72          // LDS row stride for 64-col KV half; 2-row offset = 144 -> +16 banks

__device__ __forceinline__ float phi(float x) {
    // elu(x) + 1 + 1e-6 ;  x>0: x + 1 + eps ; x<=0: expm1(x)+1+eps = exp(x)+eps
    return x > 0.0f ? (x + 1.000001f) : (__expf(x) + 1e-6f);
}

// Async DMA: 16 bytes per lane, from global address sbase plus 32-bit offset voff to LDS address lds_addr. ASYNCcnt-tracked.
__device__ __forceinline__ void async_ld16(uint32_t lds_addr, const void* sbase, uint32_t voff) {
    asm volatile("global_load_async_to_lds_b128 %0, %1, %2"
                 :: "v"(lds_addr), "v"(voff), "s"(sbase)
                 : "memory");
}
__device__ __forceinline__ void wait_async0() {
    asm volatile("s_wait_asynccnt 0x0" ::: "memory");
}

__global__ __launch_bounds__(256) void la_zero_kernel(float* ws, int total) {
    int i = blockIdx.x * 256 + threadIdx.x;
    if (i < total) ws[i] = 0.0f;
}

// ---------------- Phase 1: KV[d,e] += phi(k)^T v, ksum[d] += phi(k) ----------------
__global__ __launch_bounds__(256) void la_phase1_kernel(const float* __restrict__ k,
                                                        const float* __restrict__ v,
                                                        float* __restrict__ kv_acc,
                                                        float* __restrict__ ksum_acc) {
    __shared__ __align__(16) float skp[2][NSTEP * P1_STR];   // raw k   [16][136] x2
    __shared__ __align__(16) float svs[2][NSTEP * P1_STR];   // raw v   [16][136] x2

    const int head  = blockIdx.y;
    const int tid   = threadIdx.x;
    const int lane  = tid & 31;
    const int wave  = tid >> 5;                 // d-tile 0..7
    const int m     = lane & 15;                // M (A) / N (B) index
    const int kb    = (lane >> 4) << 1;         // K base: 0 or 2
    const int d0    = wave * 16;
    const long base = (long)head * SEQ * DIM;
    const float* kbase = k + base;
    const float* vbase = v + base;
    const int n_begin = blockIdx.x * CHUNK;

    const uint32_t skp_base = (uint32_t)(uintptr_t)&skp[0][0];
    const uint32_t svs_base = (uint32_t)(uintptr_t)&svs[0][0];
    const uint32_t buf_bytes = (uint32_t)(NSTEP * P1_STR * sizeof(float));

    // issue one stage: 16 x 128 floats of k and v, 1 float4 per thread per iter
    auto issue_stage = [&](int s) {
        const int n0  = n_begin + s * NSTEP;
        const uint32_t boff = (uint32_t)(s & 1) * buf_bytes;
        #pragma unroll
        for (int i = 0; i < 2; ++i) {
            int flat4 = tid + i * 256;              // 0..511
            int row   = flat4 >> 5;                 // 0..15
            int col   = (flat4 & 31) << 2;          // 0..124
            uint32_t goff = (uint32_t)(((n0 + row) * DIM + col) * sizeof(float));
            uint32_t loff = (uint32_t)((row * P1_STR + col) * sizeof(float));
            async_ld16(skp_base + boff + loff, kbase, goff);
            async_ld16(svs_base + boff + loff, vbase, goff);
        }
    };

    v8f acc[8] = {};                            // 8 e-tiles of 16x16 f32
    float ksum_local = 0.0f;                    // valid for tid < 128 (column tid)

    issue_stage(0);
    for (int s = 0; s < NSTAGES; ++s) {
        wait_async0();        // stage s resident in LDS (this wave's DMAs)
        __syncthreads();      // visible to all waves; buffer (s+1)&1 free of readers
        if (s + 1 < NSTAGES) issue_stage(s + 1);   // DMA next stage under the WMMAs

        const float* kp = skp[s & 1];
        const float* vp = svs[s & 1];

        // ksum partial: thread t<128 owns column t
        if (tid < 128) {
            float sum = 0.0f;
            #pragma unroll
            for (int r = 0; r < NSTEP; ++r) sum += phi(kp[r * P1_STR + tid]);
            ksum_local += sum;
        }
        // WMMA: A = phi(k)^T tile (d x n), B = raw v tile (n x e)
        #pragma unroll
        for (int kk = 0; kk < NSTEP; kk += 4) {
            v2f a;
            a.x = phi(kp[(kk + kb + 0) * P1_STR + d0 + m]);
            a.y = phi(kp[(kk + kb + 1) * P1_STR + d0 + m]);
            #pragma unroll
            for (int et = 0; et < 8; ++et) {
                v2f b;
                b.x = vp[(kk + kb + 0) * P1_STR + et * 16 + m];
                b.y = vp[(kk + kb + 1) * P1_STR + et * 16 + m];
                acc[et] = __builtin_amdgcn_wmma_f32_16x16x4_f32(
                    false, a, false, b, (short)0, acc[et], false, false);
            }
        }
    }

    // reduce partial KV into global accumulator
    const int mrow = (lane >> 4) << 3;          // 0 or 8
    #pragma unroll
    for (int et = 0; et < 8; ++et)
        #pragma unroll
        for (int r = 0; r < 8; ++r) {
            int row_d = d0 + mrow + r;
            int col_e = et * 16 + m;
            atomicAdd(&kv_acc[head * (DIM * DIM) + row_d * DIM + col_e], acc[et][r]);
        }
    if (tid < 128) atomicAdd(&ksum_acc[head * DIM + tid], ksum_local);
}

// ---------------- Phase 2: out = (phi(q) @ KV) / (phi(q) . ksum) ----------------
__global__ __launch_bounds__(256) void la_phase2_kernel(const float* __restrict__ q,
                                                        const float* __restrict__ kv_acc,
                                                        const float* __restrict__ ksum_acc,
                                                        float* __restrict__ out) {
    __shared__ __align__(16) float skv[DIM * P2_STR];   // 128 x 64 KV half, padded
    __shared__ float sksum[DIM];
    __shared__ float snorm[128];

    const int head  = blockIdx.y;
    const int rb    = blockIdx.x;               // row block: 128 q rows
    const int half  = blockIdx.z;               // e half: 0 or 1
    const int tid   = threadIdx.x;
    const int lane  = tid & 31;
    const int wave  = tid >> 5;
    const int m     = lane & 15;
    const int kb    = (lane >> 4) << 1;
    const long base = (long)head * SEQ * DIM;

    // async-stage KV half (128 x 64) into LDS
    {
        const float* gsrc = kv_acc + head * (DIM * DIM) + half * 64;
        const uint32_t lbase = (uint32_t)(uintptr_t)&skv[0];
        #pragma unroll
        for (int i = 0; i < 8; ++i) {
            int flat4 = tid + i * 256;          // 0..2047
            int d     = flat4 >> 4;             // 16 float4 per 64-col row
            int col   = (flat4 & 15) << 2;
            uint32_t goff = (uint32_t)((d * DIM + col) * sizeof(float));
            uint32_t loff = (uint32_t)((d * P2_STR + col) * sizeof(float));
            async_ld16(lbase + loff, gsrc, goff);
        }
    }
    if (tid < DIM) sksum[tid] = ksum_acc[head * DIM + tid];
    wait_async0();
    __syncthreads();

    // per-row normalizer (v/128 and *v_length cancel -> plain 1/s)
    if (tid < 128) {
        const float* qrow = q + base + (long)(rb * 128 + tid) * DIM;
        float s = 0.0f;
        #pragma unroll 8
        for (int d = 0; d < DIM; ++d) s += phi(qrow[d]) * sksum[d];
        snorm[tid] = 1.0f / s;
    }
    __syncthreads();

    // wave owns 16 q rows x 64 e cols (4 e-tiles)
    const int row0 = rb * 128 + wave * 16;
    v8f acc[4] = {};
    for (int d = 0; d < DIM; d += 4) {
        const float* qp = q + base + (long)(row0 + m) * DIM + d + kb;
        v2f a;
        a.x = phi(qp[0]);
        a.y = phi(qp[1]);
        #pragma unroll
        for (int et = 0; et < 4; ++et) {
            v2f b;
            b.x = skv[(d + kb + 0) * P2_STR + et * 16 + m];
            b.y = skv[(d + kb + 1) * P2_STR + et * 16 + m];
            acc[et] = __builtin_amdgcn_wmma_f32_16x16x4_f32(
                false, a, false, b, (short)0, acc[et], false, false);
        }
    }

    const int mrow = (lane >> 4) << 3;          // 0 or 8
    #pragma unroll
    for (int et = 0; et < 4; ++et)
        #pragma unroll
        for (int r = 0; r < 8; ++r) {
            int rl  = wave * 16 + mrow + r;     // local row in block (0..127)
            long row = (long)rb * 128 + rl;
            out[base + row * DIM + half * 64 + et * 16 + m] = acc[et][r] * snorm[rl];
        }
}

extern "C" void kernel_launch(void* const* d_in, const int* in_sizes, int n_in,
                              void* d_out, int out_size, void* d_ws, size_t ws_size,
                              hipStream_t stream) {
    const float* q = (const float*)d_in[0];
    const float* k = (const float*)d_in[1];
    const float* v = (const float*)d_in[2];
    float* out = (float*)d_out;

    float* kv_acc   = (float*)d_ws;                       // 64 * 128 * 128 floats
    float* ksum_acc = kv_acc + (size_t)HEADS * DIM * DIM; // 64 * 128 floats
    const int total = HEADS * DIM * DIM + HEADS * DIM;

    la_zero_kernel<<<(total + 255) / 256, 256, 0, stream>>>(kv_acc, total);
    la_phase1_kernel<<<dim3(SEQ / CHUNK, HEADS), 256, 0, stream>>>(k, v, kv_acc, ksum_acc);
    la_phase2_kernel<<<dim3(SEQ / 128, HEADS, 2), 256, 0, stream>>>(q, kv_acc, ksum_acc, out);
}